// PDFA_16982300688421
// MI455X (gfx1250) — compile-verified
//
#include <hip/hip_runtime.h>
#include <hip/hip_bf16.h>

// Problem constants (from reference)
#define B_    16
#define L_    2048
#define C_    16
#define S_    64
#define SEG   64
#define NSEG  (L_ / SEG)     // 32
#define PITCH 68             // LDS row pitch: 16B-aligned v4f rows, bank-conflict-free frags

typedef __attribute__((ext_vector_type(2))) float v2f;
typedef __attribute__((ext_vector_type(4))) float v4f;
typedef __attribute__((ext_vector_type(8))) float v8f;

// Workspace layout (float offsets)
#define WS_T     0                       // 65536 floats: softmaxed transition probs (S,C,S)
#define WS_INIT  (S_ * C_ * S_)          // 64 floats: softmaxed initial state probs
#define WS_ACC   (WS_INIT + S_)          // 64 floats: log-sigmoid accepting probs
#define WS_SEGP  (WS_ACC + S_ + 64)      // B*NSEG*S*S floats: per-segment matrix products

// ---------------------------------------------------------------------------
// Kernel 1: row-wise softmax of transition logits (S*C rows of length S)
// ---------------------------------------------------------------------------
__global__ void pdfa_softmax_rows(const float* __restrict__ logits,
                                  float* __restrict__ out) {
    __shared__ float red[S_];
    const int r = blockIdx.x, t = threadIdx.x;
    float x = logits[r * S_ + t];
    red[t] = x; __syncthreads();
    for (int o = 32; o > 0; o >>= 1) { if (t < o) red[t] = fmaxf(red[t], red[t + o]); __syncthreads(); }
    float mx = red[0]; __syncthreads();
    float e = __expf(x - mx);
    red[t] = e; __syncthreads();
    for (int o = 32; o > 0; o >>= 1) { if (t < o) red[t] += red[t + o]; __syncthreads(); }
    out[r * S_ + t] = e / red[0];
}

// ---------------------------------------------------------------------------
// Kernel 2: softmax(initial logits) and log_sigmoid(accepting logits), 64 lanes
// ---------------------------------------------------------------------------
__global__ void pdfa_vec_prep(const float* __restrict__ init_logits,
                              const float* __restrict__ acc_logits,
                              float* __restrict__ init_out,
                              float* __restrict__ acc_out) {
    __shared__ float red[S_];
    const int t = threadIdx.x;
    float x = init_logits[t];
    red[t] = x; __syncthreads();
    for (int o = 32; o > 0; o >>= 1) { if (t < o) red[t] = fmaxf(red[t], red[t + o]); __syncthreads(); }
    float mx = red[0]; __syncthreads();
    float e = __expf(x - mx);
    red[t] = e; __syncthreads();
    for (int o = 32; o > 0; o >>= 1) { if (t < o) red[t] += red[t + o]; __syncthreads(); }
    init_out[t] = e / red[0];
    float a = acc_logits[t];
    acc_out[t] = fminf(a, 0.0f) - log1pf(__expf(-fabsf(a)));
}

// ---------------------------------------------------------------------------
// Kernel 3: per-(batch,segment) ordered product of SEG step matrices.
//   P <- I;  for l ascending: P <- P @ M_l,  M_l = sum_c exp(logp[b,l,c]) T_c
// 512 threads (16 waves). T slices live in registers (128 floats/thread);
// M and P are double-buffered in LDS; one barrier per step (software pipeline:
// WMMA step s  ||  form M_{s+1} from registers  ||  exp weights for s+2).
// Each wave owns one 16x16 output tile; K accumulated in two WMMA chains.
// ---------------------------------------------------------------------------
__global__ __launch_bounds__(512) void pdfa_segprod(
        const float* __restrict__ logp,   // (B,L,C)
        const float* __restrict__ T,      // (S,C,S) probabilities (ws)
        float* __restrict__ segP)         // (B,NSEG,S,S)
{
    extern __shared__ float smem[];
    float* sM0 = smem;                      // S*PITCH
    float* sM1 = sM0 + S_ * PITCH;          // S*PITCH
    float* sP0 = sM1 + S_ * PITCH;          // S*PITCH
    float* sP1 = sP0 + S_ * PITCH;          // S*PITCH
    float* sW0 = sP1 + S_ * PITCH;          // 16 weights (double-buffered)
    float* sW1 = sW0 + 32;

    const int t    = threadIdx.x;
    const int seg  = blockIdx.x;
    const int b    = blockIdx.y;
    const int wave = t >> 5;
    const int lane = t & 31;
    const int lr   = lane & 15;   // row/col within tile
    const int lh   = lane >> 4;   // lane-half: K sub-pair / row-half
    const int ti   = wave >> 2;   // output tile row
    const int tj   = wave & 3;    // output tile col

    // --- T slice into registers: this thread owns M[myp][myn0..myn0+7] ---
    const int myp  = t >> 3;            // 0..63
    const int myn0 = (t & 7) << 3;      // 0,8,...,56
    v4f tr[C_][2];
    {
        const v4f* g4 = (const v4f*)(T + myp * (C_ * S_) + myn0);
        #pragma unroll
        for (int c = 0; c < C_; ++c) {
            tr[c][0] = g4[c * (S_ / 4) + 0];
            tr[c][1] = g4[c * (S_ / 4) + 1];
        }
    }

    // --- P = identity; weights for steps 0 and 1 ---
    #pragma unroll
    for (int e = 0; e < 8; ++e) {
        int idx = t * 8 + e, r = idx >> 6, c = idx & 63;
        sP0[r * PITCH + c] = (r == c) ? 1.0f : 0.0f;
    }
    const float* lpb = logp + ((size_t)b * L_ + (size_t)seg * SEG) * C_;
    if (t < C_) {
        sW0[t] = __expf(lpb[0 * C_ + t]);
        sW1[t] = __expf(lpb[1 * C_ + t]);
    }
    __syncthreads();

    // --- form M_0 into sM0 ---
    {
        v4f m0 = {}, m1 = {};
        #pragma unroll
        for (int c = 0; c < C_; ++c) {
            float w = sW0[c];
            m0 += w * tr[c][0];
            m1 += w * tr[c][1];
        }
        v4f* dst = (v4f*)(sM0 + myp * PITCH + myn0);
        dst[0] = m0; dst[1] = m1;
    }
    __syncthreads();

    float* Pc = sP0;
    float* Pn = sP1;
    const int coln = tj * 16 + lr;

    for (int s = 0; s < SEG; ++s) {
        float* Mc = (s & 1) ? sM1 : sM0;
        float* Mn = (s & 1) ? sM0 : sM1;
        float* Wn = ((s + 1) & 1) ? sW1 : sW0;   // weights for step s+1
        float* Wf = ((s + 2) & 1) ? sW1 : sW0;   // slot for step s+2

        // ---- WMMA: tile(ti,tj) of P_{s+1} = P_s @ M_s (two K chains) ----
        v8f accE = {}, accO = {};
        const float* Arow = Pc + (ti * 16 + lr) * PITCH;
        #pragma unroll
        for (int k = 0; k < 16; k += 2) {
            int ka = k * 4 + lh * 2;
            v2f a  = *(const v2f*)(Arow + ka);
            v2f bb; bb.x = Mc[ka * PITCH + coln]; bb.y = Mc[(ka + 1) * PITCH + coln];
            accE = __builtin_amdgcn_wmma_f32_16x16x4_f32(
                false, a, false, bb, (short)0, accE, false, false);
            int kb = ka + 4;
            v2f a2  = *(const v2f*)(Arow + kb);
            v2f bb2; bb2.x = Mc[kb * PITCH + coln]; bb2.y = Mc[(kb + 1) * PITCH + coln];
            accO = __builtin_amdgcn_wmma_f32_16x16x4_f32(
                false, a2, false, bb2, (short)0, accO, false, false);
        }
        v8f acc = accE + accO;

        // ---- overlap: form M_{s+1} from register-resident T ----
        if (s + 1 < SEG) {
            v4f m0 = {}, m1 = {};
            v4f wv0 = ((const v4f*)Wn)[0], wv1 = ((const v4f*)Wn)[1];
            v4f wv2 = ((const v4f*)Wn)[2], wv3 = ((const v4f*)Wn)[3];
            #pragma unroll
            for (int c = 0; c < 4; ++c) {
                m0 += wv0[c] * tr[c][0];       m1 += wv0[c] * tr[c][1];
                m0 += wv1[c] * tr[c + 4][0];   m1 += wv1[c] * tr[c + 4][1];
                m0 += wv2[c] * tr[c + 8][0];   m1 += wv2[c] * tr[c + 8][1];
                m0 += wv3[c] * tr[c + 12][0];  m1 += wv3[c] * tr[c + 12][1];
            }
            v4f* dst = (v4f*)(Mn + myp * PITCH + myn0);
            dst[0] = m0; dst[1] = m1;
        }
        // ---- overlap: weights for step s+2 ----
        if (t < C_ && s + 2 < SEG) {
            Wf[t] = __expf(lpb[(s + 2) * C_ + t]);
            __builtin_prefetch(lpb + (s + 8 < SEG ? (s + 8) : s) * C_, 0, 1);
        }

        // ---- write P_{s+1} tile to the other buffer ----
        #pragma unroll
        for (int v = 0; v < 8; ++v) {
            int r = ti * 16 + v + lh * 8;
            Pn[r * PITCH + coln] = acc[v];
        }
        float* tmp = Pc; Pc = Pn; Pn = tmp;
        __syncthreads();
    }

    // --- store segment product densely (pitch removed) ---
    float* outp = segP + ((size_t)(b * NSEG + seg) << 12);
    #pragma unroll
    for (int e = 0; e < 8; ++e) {
        int idx = t * 8 + e, r = idx >> 6, c = idx & 63;
        outp[idx] = Pc[r * PITCH + c];
    }
}

// ---------------------------------------------------------------------------
// Kernel 4: per-batch vector chain over segment products + logsumexp epilogue.
// ---------------------------------------------------------------------------
__global__ void pdfa_combine(const float* __restrict__ segP,
                             const float* __restrict__ initp,
                             const float* __restrict__ accLP,
                             float* __restrict__ out) {
    __shared__ float v[S_];
    __shared__ float red[S_];
    const int b = blockIdx.x, t = threadIdx.x;
    v[t] = initp[t];
    __syncthreads();
    for (int seg = 0; seg < NSEG; ++seg) {
        const float* P = segP + ((size_t)(b * NSEG + seg) << 12);
        float s = 0.0f;
        #pragma unroll 8
        for (int p = 0; p < S_; ++p)
            s = fmaf(v[p], P[p * S_ + t], s);
        __syncthreads();
        v[t] = s;
        __syncthreads();
    }
    float lp = logf(v[t] + 1e-20f) + accLP[t];
    red[t] = lp; __syncthreads();
    for (int o = 32; o > 0; o >>= 1) { if (t < o) red[t] = fmaxf(red[t], red[t + o]); __syncthreads(); }
    float mx = red[0]; __syncthreads();
    red[t] = __expf(lp - mx); __syncthreads();
    for (int o = 32; o > 0; o >>= 1) { if (t < o) red[t] += red[t + o]; __syncthreads(); }
    if (t == 0) out[b] = mx + logf(red[0]);
}

// ---------------------------------------------------------------------------
extern "C" void kernel_launch(void* const* d_in, const int* in_sizes, int n_in,
                              void* d_out, int out_size, void* d_ws, size_t ws_size,
                              hipStream_t stream) {
    const float* logp         = (const float*)d_in[0];  // (B,L,C)
    const float* init_logits  = (const float*)d_in[1];  // (S,)
    const float* trans_logits = (const float*)d_in[2];  // (S,C,S)
    const float* acc_logits   = (const float*)d_in[3];  // (S,)
    float* ws     = (float*)d_ws;
    float* wsT    = ws + WS_T;
    float* wsInit = ws + WS_INIT;
    float* wsAcc  = ws + WS_ACC;
    float* wsSeg  = ws + WS_SEGP;
    float* outp   = (float*)d_out;                      // (1,B) -> 16 floats

    pdfa_softmax_rows<<<S_ * C_, S_, 0, stream>>>(trans_logits, wsT);
    pdfa_vec_prep<<<1, S_, 0, stream>>>(init_logits, acc_logits, wsInit, wsAcc);

    const size_t smem = (size_t)(4 * S_ * PITCH + 64) * sizeof(float);
    hipFuncSetAttribute(reinterpret_cast<const void*>(pdfa_segprod),
                        hipFuncAttributeMaxDynamicSharedMemorySize, (int)smem);
    pdfa_segprod<<<dim3(NSEG, B_), 512, smem, stream>>>(logp, wsT, wsSeg);

    pdfa_combine<<<B_, S_, 0, stream>>>(wsSeg, wsInit, wsAcc, outp);
}